// RNNModel_89593017795141
// MI455X (gfx1250) — compile-verified
//
#include <hip/hip_runtime.h>

typedef __attribute__((ext_vector_type(2))) float v2f;
typedef __attribute__((ext_vector_type(8))) float v8f;

#define RNN_B 8192
#define RNN_T 512
#define RNN_I 9
#define RNN_H 6
#define TILES_PER_WAVE 8

// tanh(v) = 1 - 2/(exp(2v)+1); exp(2v) = exp2(2v*log2(e)).
// Saturates correctly for large |v|.
__device__ __forceinline__ float fast_tanhf(float v) {
  float e = __builtin_amdgcn_exp2f(v * 2.88539008177792681472f); // 2*log2(e)
  return 1.0f - 2.0f * __builtin_amdgcn_rcpf(e + 1.0f);
}

// ---------------------------------------------------------------------------
// Kernel 1: xp[t][b][0..5] = x[b][t][:] @ W_ih^T + (b_ih + b_hh)
// V_WMMA_F32_16X16X4_F32, K padded 9->12, N padded 6->16.
// All padding via clamped-address loads + selects (no EXEC divergence before
// the WMMAs). Each wave processes 8 consecutive 16-row tiles so the B-matrix
// and bias registers are loaded once per wave.
// xp laid out [T, B, 8] (8-float rows, 32B aligned) for the scan kernel.
// ---------------------------------------------------------------------------
__global__ __launch_bounds__(256) void xproj_wmma_f32(
    const float* __restrict__ x, const float* __restrict__ W_ih,
    const float* __restrict__ b_ih, const float* __restrict__ b_hh,
    float* __restrict__ xp)
{
  const int  lane = threadIdx.x & 31;
  const int  l15  = lane & 15;
  const int  half = lane >> 4;
  const long wave = ((long)blockIdx.x * blockDim.x + threadIdx.x) >> 5;
  const long waveRowBase = wave * (TILES_PER_WAVE * 16);

  // ---- per-lane K indices for the 3 k-steps (constant per kernel) ----
  // k-step s covers K = 4s + half*2, 4s + half*2 + 1
  int  kIdx[6];
  bool kVal[6];
#pragma unroll
  for (int s = 0; s < 3; ++s) {
    const int kA = 4 * s + half * 2;
    kVal[2*s+0] = (kA     < RNN_I);
    kVal[2*s+1] = (kA + 1 < RNN_I);
    kIdx[2*s+0] = kVal[2*s+0] ? kA     : (RNN_I - 1);  // clamp -> valid addr
    kIdx[2*s+1] = kVal[2*s+1] ? kA + 1 : (RNN_I - 1);
  }

  // ---- B matrix (W_ih^T zero-padded) + bias: loaded once per wave ----
  const bool nVal   = (l15 < RNN_H);
  const int  nClamp = nVal ? l15 : (RNN_H - 1);

  float bi = b_ih[nClamp];
  float bh = b_hh[nClamp];
  const float bias = nVal ? (bi + bh) : 0.0f;

  v2f bm[3];
#pragma unroll
  for (int s = 0; s < 3; ++s) {
    float w0 = W_ih[nClamp * RNN_I + kIdx[2*s+0]];   // unconditional load
    float w1 = W_ih[nClamp * RNN_I + kIdx[2*s+1]];
    bm[s].x = (nVal && kVal[2*s+0]) ? w0 : 0.0f;     // v_cndmask
    bm[s].y = (nVal && kVal[2*s+1]) ? w1 : 0.0f;
  }

  // ---- tile loop ----
#pragma unroll 1
  for (int tt = 0; tt < TILES_PER_WAVE; ++tt) {
    const long rowBase = waveRowBase + (long)tt * 16;
    const float* xrow = x + (rowBase + l15) * RNN_I;   // A row for lane m=l15

    v8f c;
#pragma unroll
    for (int j = 0; j < 8; ++j) c[j] = bias;

#pragma unroll
    for (int s = 0; s < 3; ++s) {
      float r0 = xrow[kIdx[2*s+0]];                    // unconditional loads
      float r1 = xrow[kIdx[2*s+1]];
      v2f a;
      a.x = kVal[2*s+0] ? r0 : 0.0f;
      a.y = kVal[2*s+1] ? r1 : 0.0f;
      c = __builtin_amdgcn_wmma_f32_16x16x4_f32(false, a, false, bm[s],
                                                (short)0, c, false, false);
    }

    // C layout: VGPR j = row (j + 8*half), column n = l15.
    // Store columns 0..7 (6,7 are exact zeros) to xp[(t*B + b)*8 + n].
    if (l15 < 8) {
#pragma unroll
      for (int j = 0; j < 8; ++j) {
        long r  = rowBase + j + half * 8;   // global row = b*T + t
        long bb = r >> 9;                   // / T  (T = 512)
        long ti = r & (RNN_T - 1);
        xp[(ti * (long)RNN_B + bb) * 8 + l15] = c[j];
      }
    }
  }
}

// ---------------------------------------------------------------------------
// Kernel 2: sequential scan over T, one thread per batch row.
// h_t = tanh(xp_t + W_hh h_{t-1}); out = W_fc h_T + b_fc.
// Software-pipelined next-step load + prefetch 8 steps ahead; [T,B,8] layout
// means one wave reads a contiguous 1 KB per step.
// ---------------------------------------------------------------------------
__global__ __launch_bounds__(256) void rnn_scan(
    const float* __restrict__ xp, const float* __restrict__ W_hh,
    const float* __restrict__ W_fc, const float* __restrict__ b_fc,
    float* __restrict__ out)
{
  const int b = blockIdx.x * blockDim.x + threadIdx.x;

  float w[RNN_H][RNN_H];
#pragma unroll
  for (int j = 0; j < RNN_H; ++j)
#pragma unroll
    for (int i = 0; i < RNN_H; ++i) w[j][i] = W_hh[j * RNN_H + i];

  float h[RNN_H];
#pragma unroll
  for (int j = 0; j < RNN_H; ++j) h[j] = 0.0f;

  const float4* base = (const float4*)xp;   // float4 index = (t*B + b)*2
  float4 cur0 = base[(long)b * 2 + 0];
  float4 cur1 = base[(long)b * 2 + 1];

  for (int t = 0; t < RNN_T; ++t) {
    const int  tn = (t + 1 < RNN_T) ? (t + 1) : (RNN_T - 1);
    const long ni = ((long)tn * RNN_B + b) * 2;
    float4 nx0 = base[ni];
    float4 nx1 = base[ni + 1];
    const int tp = (t + 8 < RNN_T) ? (t + 8) : (RNN_T - 1);
    __builtin_prefetch(&base[((long)tp * RNN_B + b) * 2], 0, 0);

    const float xv[RNN_H] = {cur0.x, cur0.y, cur0.z, cur0.w, cur1.x, cur1.y};
    float nh[RNN_H];
#pragma unroll
    for (int j = 0; j < RNN_H; ++j) {
      float acc = xv[j];
#pragma unroll
      for (int i = 0; i < RNN_H; ++i) acc = fmaf(w[j][i], h[i], acc);
      nh[j] = fast_tanhf(acc);
    }
#pragma unroll
    for (int j = 0; j < RNN_H; ++j) h[j] = nh[j];
    cur0 = nx0; cur1 = nx1;
  }

  float acc = b_fc[0];
#pragma unroll
  for (int j = 0; j < RNN_H; ++j) acc = fmaf(W_fc[j], h[j], acc);
  out[b] = acc;
}

// ---------------------------------------------------------------------------
// Fallback (workspace too small): fully fused, reads x directly.
// ---------------------------------------------------------------------------
__global__ __launch_bounds__(256) void rnn_fused(
    const float* __restrict__ x, const float* __restrict__ W_ih,
    const float* __restrict__ W_hh, const float* __restrict__ b_ih,
    const float* __restrict__ b_hh, const float* __restrict__ W_fc,
    const float* __restrict__ b_fc, float* __restrict__ out)
{
  const int b = blockIdx.x * blockDim.x + threadIdx.x;

  float wih[RNN_H][RNN_I], whh[RNN_H][RNN_H], bias[RNN_H];
#pragma unroll
  for (int j = 0; j < RNN_H; ++j) {
    bias[j] = b_ih[j] + b_hh[j];
#pragma unroll
    for (int i = 0; i < RNN_I; ++i) wih[j][i] = W_ih[j * RNN_I + i];
#pragma unroll
    for (int i = 0; i < RNN_H; ++i) whh[j][i] = W_hh[j * RNN_H + i];
  }

  float h[RNN_H];
#pragma unroll
  for (int j = 0; j < RNN_H; ++j) h[j] = 0.0f;

  const float* xr = x + (long)b * RNN_T * RNN_I;
  for (int t = 0; t < RNN_T; ++t) {
    float xv[RNN_I];
#pragma unroll
    for (int i = 0; i < RNN_I; ++i) xv[i] = xr[(long)t * RNN_I + i];
    float nh[RNN_H];
#pragma unroll
    for (int j = 0; j < RNN_H; ++j) {
      float acc = bias[j];
#pragma unroll
      for (int i = 0; i < RNN_I; ++i) acc = fmaf(wih[j][i], xv[i], acc);
#pragma unroll
      for (int i = 0; i < RNN_H; ++i) acc = fmaf(whh[j][i], h[i], acc);
      nh[j] = fast_tanhf(acc);
    }
#pragma unroll
    for (int j = 0; j < RNN_H; ++j) h[j] = nh[j];
  }

  float acc = b_fc[0];
#pragma unroll
  for (int j = 0; j < RNN_H; ++j) acc = fmaf(W_fc[j], h[j], acc);
  out[b] = acc;
}

extern "C" void kernel_launch(void* const* d_in, const int* in_sizes, int n_in,
                              void* d_out, int out_size, void* d_ws, size_t ws_size,
                              hipStream_t stream) {
  const float* x    = (const float*)d_in[0];
  const float* W_ih = (const float*)d_in[1];
  const float* W_hh = (const float*)d_in[2];
  const float* b_ih = (const float*)d_in[3];
  const float* b_hh = (const float*)d_in[4];
  const float* W_fc = (const float*)d_in[5];
  const float* b_fc = (const float*)d_in[6];
  float* out = (float*)d_out;

  const size_t need = (size_t)RNN_T * RNN_B * 8 * sizeof(float); // 128 MiB
  if (ws_size >= need) {
    float* xp = (float*)d_ws;
    // total tiles = B*T/16 = 262144; 8 tiles per wave -> 32768 waves
    const long waves  = ((long)RNN_B * RNN_T / 16) / TILES_PER_WAVE;
    const int  blocks = (int)(waves * 32 / 256);   // 4096
    xproj_wmma_f32<<<dim3(blocks), dim3(256), 0, stream>>>(x, W_ih, b_ih, b_hh, xp);
    rnn_scan<<<dim3(RNN_B / 256), dim3(256), 0, stream>>>(xp, W_hh, W_fc, b_fc, out);
  } else {
    rnn_fused<<<dim3(RNN_B / 256), dim3(256), 0, stream>>>(x, W_ih, W_hh, b_ih,
                                                           b_hh, W_fc, b_fc, out);
  }
}